// C_idht_60215441490183
// MI455X (gfx1250) — compile-verified
//
#include <hip/hip_runtime.h>

// ---------------------------------------------------------------------------
// Inverse Deep Hough Transform for MI455X (gfx1250)
//   out[n,c,h,w] = sum_a acc[n,c,a, round(x*cos+y*sin)+200]
// One workgroup per (n,c) slice; angle chunks double-buffered in LDS via the
// CDNA5 async global->LDS engine (ASYNCcnt), gathers served from LDS.
// For H=W=256, r is provably always in [18,382] -> no validity mask needed.
// ---------------------------------------------------------------------------

#define OUT_H       256
#define OUT_W       256
#define NUMANGLE    180
#define NUMRHO      400
#define NSLICE      256      // N*C = 4*64
#define CHUNK_A     20       // angles per LDS chunk
#define NCHUNK      9        // 180 / 20
#define THREADS     1024     // 32 waves (wave32)
#define ROWS_PER_T  64       // 65536 pixels / 1024 threads

// 16-byte int vector in the address spaces the async-LDS builtin expects.
typedef int v4i __attribute__((vector_size(16)));
typedef __attribute__((address_space(1))) v4i g_v4i;   // global (__device__)
typedef __attribute__((address_space(3))) v4i l_v4i;   // LDS    (__shared__)
typedef __attribute__((address_space(3))) float lds_f32_t;

// ---- CDNA5 async global->LDS copy (ASYNCcnt-tracked) ----------------------
__device__ __forceinline__ void async_copy_b128(const float* g, float* l) {
#if __has_builtin(__builtin_amdgcn_global_load_async_to_lds_b128)
    __builtin_amdgcn_global_load_async_to_lds_b128(
        (g_v4i*)(const g_v4i*)g, (l_v4i*)l, /*offset=*/0, /*cpol=*/0);
#else
    unsigned loff = (unsigned)(__UINTPTR_TYPE__)(lds_f32_t*)l;
    asm volatile("global_load_async_to_lds_b128 %0, %1, off"
                 :: "v"(loff), "v"(g) : "memory");
#endif
}

__device__ __forceinline__ void wait_async0() {
#if __has_builtin(__builtin_amdgcn_s_wait_asynccnt)
    __builtin_amdgcn_s_wait_asynccnt(0);
#else
    asm volatile("s_wait_asynccnt 0" ::: "memory");
#endif
}

// ---------------------------------------------------------------------------

__global__ __launch_bounds__(THREADS, 1)
void idht_accum_kernel(const float* __restrict__ acc, float* __restrict__ out) {
    __shared__ float s_buf[2][CHUNK_A * NUMRHO];   // 2 * 32000 B = 64000 B
    __shared__ float s_cos[NUMANGLE];              // 720 B
    __shared__ float s_sin[NUMANGLE];              // 720 B  (total 65440 B)

    const int nc = blockIdx.x;                     // (n*C + c) slice id
    const int t  = threadIdx.x;

    const float* slice = acc + (size_t)nc * (NUMANGLE * NUMRHO);

    // FIRST: kick off the async DMA of chunk 0 so it overlaps the
    // transcendental table build below. (global_load_async_to_lds_b128)
#pragma unroll
    for (int k = 0; k < (CHUNK_A * NUMRHO) / 4 / THREADS + 1; ++k) {
        int i = k * THREADS + t;
        if (i < (CHUNK_A * NUMRHO) / 4)
            async_copy_b128(slice + 4 * i, &s_buf[0][4 * i]);
    }

    // Angle tables (one-time, tiny); overlapped with the DMA above.
    if (t < NUMANGLE) {
        float ang = (float)t * (3.14159265358979323846f / (float)NUMANGLE);
        float sv, cv;
        sincosf(ang, &sv, &cv);
        s_sin[t] = sv;
        s_cos[t] = cv;
    }

    const int w  = t & (OUT_W - 1);                // column owned by thread
    const int q  = t >> 8;                         // row quadrant 0..3
    const int h0 = q * ROWS_PER_T;

    float sum[ROWS_PER_T];
#pragma unroll
    for (int j = 0; j < ROWS_PER_T; ++j) sum[j] = 0.0f;

    const float xf = (float)(w - OUT_W / 2);
    const float y0 = (float)(h0 - OUT_H / 2);

    for (int ch = 0; ch < NCHUNK; ++ch) {
        wait_async0();          // my async writes into LDS have landed
        __syncthreads();        // everyone's writes have landed

        // Prefetch next chunk into the other buffer while we compute
        if (ch + 1 < NCHUNK) {
            const float* src = slice + (size_t)(ch + 1) * (CHUNK_A * NUMRHO);
            float*       dst = &s_buf[(ch + 1) & 1][0];
            for (int i = t; i < (CHUNK_A * NUMRHO) / 4; i += THREADS)
                async_copy_b128(src + 4 * i, dst + 4 * i);
        }

        const float* buf = &s_buf[ch & 1][0];
        const int    a0  = ch * CHUNK_A;

        for (int al = 0; al < CHUNK_A; ++al) {
            const float c  = s_cos[a0 + al];       // LDS broadcast
            const float s  = s_sin[a0 + al];
            const float xc = __fmul_rn(xf, c);     // hoisted per (thread,angle)
            const float* row = buf + al * NUMRHO;
#pragma unroll
            for (int j = 0; j < ROWS_PER_T; ++j) {
                // Match reference rounding: (x*cos) + (y*sin), no fma
                // contraction, then round-half-to-even (v_rndne).
                float v = __fadd_rn(xc, __fmul_rn(y0 + (float)j, s));
                int   r = (int)rintf(v) + NUMRHO / 2;   // always in [18,382]
                sum[j] += row[r];                       // ds_load_b32 gather
            }
        }
        __syncthreads();        // all reads of this buffer done before reuse
    }

    // Coalesced writeback: consecutive lanes -> consecutive w
    float* o = out + ((size_t)nc * OUT_H + h0) * OUT_W + w;
#pragma unroll
    for (int j = 0; j < ROWS_PER_T; ++j)
        o[(size_t)j * OUT_W] = sum[j];
}

// ---------------------------------------------------------------------------

extern "C" void kernel_launch(void* const* d_in, const int* in_sizes, int n_in,
                              void* d_out, int out_size, void* d_ws, size_t ws_size,
                              hipStream_t stream) {
    (void)in_sizes; (void)n_in; (void)out_size; (void)d_ws; (void)ws_size;
    const float* acc = (const float*)d_in[0];   // (4, 64, 180, 400) f32
    float*       out = (float*)d_out;           // (4, 64, 256, 256) f32

    dim3 grid(NSLICE);      // one block per (n,c) slice -> ~one per WGP
    dim3 block(THREADS);    // 32 wave32 waves
    hipLaunchKernelGGL(idht_accum_kernel, grid, block, 0, stream, acc, out);
}